// VectorFusionX_MHA_51539608322
// MI455X (gfx1250) — compile-verified
//
#include <hip/hip_runtime.h>

// VectorFusionX_MHA on MI455X (gfx1250).
// kv_len==1 => softmax==1 => ctx==v2: q/k paths are dead code. The live work is
// 4 chained f32 GEMVs (96 MB of weights, ~50 MFLOP) -> pure HBM-bandwidth bound
// (~4.1 us floor at 23.3 TB/s). Strategy: split-K GEMV waves using
// V_WMMA_F32_16X16X4_F32 as the MAC engine (full f32 precision), 4x-unrolled
// inner loop with hoisted loads + dual accumulators for memory-latency overlap,
// deterministic two-phase reduction (partials + ordered reduce).

#define EMB 2048
#define VIS 4096
#define KC  32      // K-chunks per GEMV (split-K parallelism)

typedef float v2f __attribute__((ext_vector_type(2)));
typedef float v8f __attribute__((ext_vector_type(8)));

static __device__ __forceinline__ v8f wmma_f32(v2f a, v2f b, v8f c) {
  return __builtin_amdgcn_wmma_f32_16x16x4_f32(
      /*neg_a=*/false, a, /*neg_b=*/false, b,
      /*c_mod=*/(short)0, c, /*reuse_a=*/false, /*reuse_b=*/false);
}

// One wave computes a 16-row x (K/KC) partial GEMV tile via WMMA f32 16x16x4.
// A (16x4) = W tile:  lane L<16 holds row L, K={k,k+1}; lane L>=16 holds row
// L-16, K={k+2,k+3}  (per CDNA5 ISA 32-bit A layout).
// B (4x16) = x chunk broadcast to all 16 columns (rows striped across lanes:
// lanes<16 hold K={k,k+1}, lanes>=16 hold K={k+2,k+3}) -> every D column holds
// the same 16x1 partial, so extraction needs no cross-lane traffic.
// D (16x16): VGPR r = M=r (lanes 0-15) / M=8+r (lanes 16-31).
__global__ __launch_bounds__(128) void gemv16_wmma(
    const float* __restrict__ W,      // M x K, row-major
    const float* __restrict__ x,      // K
    float* __restrict__ partial,      // KC x M
    int M, int K) {
  const int waveInBlk = threadIdx.x >> 5;
  const int lane      = threadIdx.x & 31;
  const int tile      = blockIdx.x * 4 + waveInBlk;   // 4 waves / block
  const int rowTile   = tile / KC;
  const int kc        = tile - rowTile * KC;
  const int kPer      = K / KC;                       // multiple of 16
  const int half      = lane >> 4;                    // 0: K-pair {0,1}, 1: {2,3}
  const int l         = lane & 15;

  const float* wp = W + (size_t)(rowTile * 16 + l) * K + kc * kPer + 2 * half;
  const float* xq = x + kc * kPer + 2 * half;

  v8f acc0 = {};
  v8f acc1 = {};
  const int iters = kPer >> 4;        // 4 WMMAs (K=16) per iteration
  for (int it = 0; it < iters; ++it) {
    // Prefetch next unroll block of the W stream (16 floats ahead per lane).
    __builtin_prefetch(wp + 16, 0, 3);
    // Hoist all loads: 8 outstanding global_load_b64 before any wait.
    v2f a0 = *(const v2f*)(wp);
    v2f a1 = *(const v2f*)(wp + 4);
    v2f a2 = *(const v2f*)(wp + 8);
    v2f a3 = *(const v2f*)(wp + 12);
    v2f b0 = *(const v2f*)(xq);
    v2f b1 = *(const v2f*)(xq + 4);
    v2f b2 = *(const v2f*)(xq + 8);
    v2f b3 = *(const v2f*)(xq + 12);
    acc0 = wmma_f32(a0, b0, acc0);
    acc1 = wmma_f32(a1, b1, acc1);
    acc0 = wmma_f32(a2, b2, acc0);
    acc1 = wmma_f32(a3, b3, acc1);
    wp += 16;
    xq += 16;
  }
  v8f acc = acc0 + acc1;

  // Column 0..15 of D are identical; lane 0 holds M=0..7, lane 16 holds M=8..15.
  if (l == 0) {
    float* p = partial + (size_t)kc * M + rowTile * 16 + half * 8;
#pragma unroll
    for (int r = 0; r < 8; ++r) p[r] = acc[r];
  }
}

// Deterministic reduction over the KC partials, plus bias (and optional
// residual term for the final stage). Fixed summation order => bit-stable.
__global__ __launch_bounds__(256) void reduce_bias(
    const float* __restrict__ partial,  // KC x M
    const float* __restrict__ bias,     // M
    const float* __restrict__ extra,    // M or nullptr (residual add)
    float* __restrict__ y, int M) {
  int i = blockIdx.x * blockDim.x + threadIdx.x;
  if (i >= M) return;
  float s = bias[i];
  if (extra) s += extra[i];
#pragma unroll
  for (int kc = 0; kc < KC; ++kc) s += partial[(size_t)kc * M + i];
  y[i] = s;
}

extern "C" void kernel_launch(void* const* d_in, const int* in_sizes, int n_in,
                              void* d_out, int out_size, void* d_ws, size_t ws_size,
                              hipStream_t stream) {
  const float* visual = (const float*)d_in[0];
  const float* text   = (const float*)d_in[1];
  // d_in[2..5] (q_w,q_b,k_w,k_b) are dead: softmax over a single kv element is 1.
  const float* v_w  = (const float*)d_in[6];
  const float* v_b  = (const float*)d_in[7];
  const float* in_w = (const float*)d_in[8];
  const float* in_b = (const float*)d_in[9];
  const float* mo_w = (const float*)d_in[10];
  const float* mo_b = (const float*)d_in[11];
  const float* o_w  = (const float*)d_in[12];
  const float* o_b  = (const float*)d_in[13];
  float* out = (float*)d_out;

  float* ws   = (float*)d_ws;
  float* v    = ws;                 // EMB floats
  float* v2   = ws + EMB;           // EMB floats
  float* attn = ws + 2 * EMB;       // EMB floats
  float* P    = ws + 4 * EMB;       // KC x VIS floats (reused per stage)

  const float* wv = in_w + (size_t)2 * EMB * EMB;  // in_w rows [2E, 3E)
  const float* bv = in_b + 2 * EMB;

  const int gE = (EMB / 16) * KC / 4;   // 1024 blocks (4096 waves)
  const int gV = (VIS / 16) * KC / 4;   // 2048 blocks (8192 waves)

  // Stage 1: v = v_w . text + v_b                  (32 MB of weights)
  gemv16_wmma<<<gE, 128, 0, stream>>>(v_w, text, P, EMB, VIS);
  reduce_bias<<<(EMB + 255) / 256, 256, 0, stream>>>(P, v_b, nullptr, v, EMB);

  // Stage 2: v2 = wv . v + bv                      (16 MB)
  gemv16_wmma<<<gE, 128, 0, stream>>>(wv, v, P, EMB, EMB);
  reduce_bias<<<(EMB + 255) / 256, 256, 0, stream>>>(P, bv, nullptr, v2, EMB);

  // Stage 3: attn = mo_w . v2 + mo_b               (16 MB)
  gemv16_wmma<<<gE, 128, 0, stream>>>(mo_w, v2, P, EMB, EMB);
  reduce_bias<<<(EMB + 255) / 256, 256, 0, stream>>>(P, mo_b, nullptr, attn, EMB);

  // Stage 4: out = visual + o_w . attn + o_b       (32 MB)
  gemv16_wmma<<<gV, 128, 0, stream>>>(o_w, attn, P, VIS, EMB);
  reduce_bias<<<(VIS + 255) / 256, 256, 0, stream>>>(P, o_b, visual, out, VIS);
}